// TransD_64828236366349
// MI455X (gfx1250) — compile-verified
//
#include <hip/hip_runtime.h>

#define BATCH   8192
#define DDIM    256
#define MARGIN  1.0f
#define EPSN    1e-12f

typedef __attribute__((ext_vector_type(2))) float v2f;
typedef __attribute__((ext_vector_type(8))) float v8f;

// ds_swizzle_b32 group-of-32 immediates: offset = xor<<10 | or<<5 | and
#define SWZ_XOR16 0x401F   // xor=0x10, and=0x1f
#define SWZ_XOR8  0x201F   // xor=0x08, and=0x1f

template <int IMM>
__device__ __forceinline__ float dswz(float x) {
    return __int_as_float(__builtin_amdgcn_ds_swizzle(__float_as_int(x), IMM));
}

__device__ __forceinline__ float dot4(float4 a, float4 b) {
    return fmaf(a.x, b.x, fmaf(a.y, b.y, fmaf(a.z, b.z, a.w * b.w)));
}

__device__ __forceinline__ float4 fma4s(float4 h, float4 rp, float s) {
    float4 o;
    o.x = fmaf(rp.x, s, h.x);
    o.y = fmaf(rp.y, s, h.y);
    o.z = fmaf(rp.z, s, h.z);
    o.w = fmaf(rp.w, s, h.w);
    return o;
}

__device__ __forceinline__ float4 comb4(float4 hh, float a, float4 r, float b,
                                        float4 tt, float c) {
    float4 o;
    o.x = fmaf(hh.x, a, fmaf(r.x, b, -tt.x * c));
    o.y = fmaf(hh.y, a, fmaf(r.y, b, -tt.y * c));
    o.z = fmaf(hh.z, a, fmaf(r.z, b, -tt.z * c));
    o.w = fmaf(hh.w, a, fmaf(r.w, b, -tt.w * c));
    return o;
}

__device__ __forceinline__ v8f wmma_f32(v2f a, v2f b) {
    v8f c = {};
    return __builtin_amdgcn_wmma_f32_16x16x4_f32(
        /*neg_a=*/false, a, /*neg_b=*/false, b,
        /*c_mod=*/(short)0, c, /*reuse_a=*/false, /*reuse_b=*/false);
}

__device__ __forceinline__ float acc_sum8(v8f c) {
    return ((c[0] + c[1]) + (c[2] + c[3])) + ((c[4] + c[5]) + (c[6] + c[7]));
}

// Single wave32 sum: A[m,2h]=p, B=ones -> D[m,*] = p_m + p_{m+16}.
// Lane<16 accs hold S_0..S_7, lane>=16 hold S_8..S_15; 7 adds + xor16.
__device__ __forceinline__ float wave_sum(float p) {
    v2f a; a[0] = p;    a[1] = 0.0f;
    v2f b; b[0] = 1.0f; b[1] = 1.0f;
    float s = acc_sum8(wmma_f32(a, b));
    return s + dswz<SWZ_XOR16>(s);
}

// Paired wave32 sums (P=sum of p, Q=sum of q) with ONE wmma.
// a[0]=p -> A[m,2h] (even K), a[1]=q -> A[m,2h+1] (odd K).
// bsel: b[0] = (n<8)?1:0 (even-K rows), b[1] = (n<8)?0:1, n = lane%16.
// => D[m, n<8] = p_m + p_{m+16}, D[m, n>=8] = q_m + q_{m+16}.
// acc-sum gives per-lane half-sums; xor16-add completes each total on its
// lane group; xor8 fetches the other total; cndmask sorts P vs Q.
__device__ __forceinline__ float2 wave_sum2(float p, float q, v2f bsel, bool low) {
    v2f a; a[0] = p; a[1] = q;
    float s = acc_sum8(wmma_f32(a, bsel));
    float own   = s + dswz<SWZ_XOR16>(s);   // Ptot on n<8 lanes, Qtot on n>=8
    float other = dswz<SWZ_XOR8>(own);
    float2 r;
    r.x = low ? own : other;   // P
    r.y = low ? other : own;   // Q
    return r;
}

__global__ __launch_bounds__(64)
void transd_loss_kernel(const int* __restrict__ X, const int* __restrict__ Y,
                        const float* __restrict__ ent_emb,
                        const float* __restrict__ rel_emb,
                        const float* __restrict__ ent_proj,
                        const float* __restrict__ rel_proj,
                        float* __restrict__ out) {
    const int b    = blockIdx.x;
    const int wid  = threadIdx.x >> 5;   // 0: positive (X), 1: negative (Y)
    const int lane = threadIdx.x & 31;

    const int* trip = (wid == 0 ? X : Y) + 3 * b;
    const int ih = trip[0];
    const int ir = trip[1];
    const int it = trip[2];

    const float4* Hrow = (const float4*)(ent_emb  + (size_t)ih * DDIM);
    const float4* Hp   = (const float4*)(ent_proj + (size_t)ih * DDIM);
    const float4* Rrow = (const float4*)(rel_emb  + (size_t)ir * DDIM);
    const float4* Rp   = (const float4*)(rel_proj + (size_t)ir * DDIM);
    const float4* Trow = (const float4*)(ent_emb  + (size_t)it * DDIM);
    const float4* Tp   = (const float4*)(ent_proj + (size_t)it * DDIM);

    const int c0 = lane;        // columns [4*lane, 4*lane+4)
    const int c1 = lane + 32;   // columns [128+4*lane, ...)

    // Issue the whole 6 KB gather up front; everything below overlaps it.
    float4 rp0 = Rp[c0],   rp1 = Rp[c1];
    float4 h0  = Hrow[c0], h1  = Hrow[c1];
    float4 hp0 = Hp[c0],   hp1 = Hp[c1];
    float4 t0  = Trow[c0], t1  = Trow[c1];
    float4 tp0 = Tp[c0],   tp1 = Tp[c1];
    float4 r0  = Rrow[c0], r1  = Rrow[c1];

    // Column-select B for paired reductions (built once).
    const bool low = (lane & 8) == 0;     // n = lane%16 < 8
    v2f bsel; bsel[0] = low ? 1.0f : 0.0f; bsel[1] = low ? 0.0f : 1.0f;

    // dh = h_p.h and dt = t_p.t in one WMMA
    float2 d2 = wave_sum2(dot4(hp0, h0) + dot4(hp1, h1),
                          dot4(tp0, t0) + dot4(tp1, t1), bsel, low);
    const float dh = d2.x, dt = d2.y;

    // nr = ||r||^2 (independent of the chain above)
    float nr = wave_sum(dot4(r0, r0) + dot4(r1, r1));

    // h_ = h + r_p*dh ; t_ = t + r_p*dt
    float4 hh0 = fma4s(h0, rp0, dh), hh1 = fma4s(h1, rp1, dh);
    float4 tt0 = fma4s(t0, rp0, dt), tt1 = fma4s(t1, rp1, dt);

    // nh = ||h_||^2 and nt = ||t_||^2 in one WMMA
    float2 n2 = wave_sum2(dot4(hh0, hh0) + dot4(hh1, hh1),
                          dot4(tt0, tt0) + dot4(tt1, tt1), bsel, low);

    // reference: x / max(||x||, eps)
    const float inh = 1.0f / fmaxf(sqrtf(n2.x), EPSN);
    const float intt = 1.0f / fmaxf(sqrtf(n2.y), EPSN);
    const float inr = 1.0f / fmaxf(sqrtf(nr), EPSN);

    // s = norm(h_) + norm(r) - norm(t_) ; loss = ||s||_2
    float4 s0 = comb4(hh0, inh, r0, inr, tt0, intt);
    float4 s1 = comb4(hh1, inh, r1, inr, tt1, intt);
    float ns = wave_sum(dot4(s0, s0) + dot4(s1, s1));
    float loss = sqrtf(ns);

    __shared__ float sl[2];
    if (lane == 0) sl[wid] = loss;
    __syncthreads();
    if (threadIdx.x == 0) {
        out[b] = fmaxf(sl[0] - sl[1] + MARGIN, 0.0f);
    }
}

extern "C" void kernel_launch(void* const* d_in, const int* in_sizes, int n_in,
                              void* d_out, int out_size, void* d_ws, size_t ws_size,
                              hipStream_t stream) {
    const int*   X        = (const int*)d_in[0];
    const int*   Y        = (const int*)d_in[1];
    const float* ent_emb  = (const float*)d_in[2];
    const float* rel_emb  = (const float*)d_in[3];
    const float* ent_proj = (const float*)d_in[4];
    const float* rel_proj = (const float*)d_in[5];
    float* out = (float*)d_out;

    transd_loss_kernel<<<BATCH, 64, 0, stream>>>(X, Y, ent_emb, rel_emb,
                                                 ent_proj, rel_proj, out);
}